// BlurModel_21277267984533
// MI455X (gfx1250) — compile-verified
//
#include <hip/hip_runtime.h>
#include <hip/hip_bf16.h>

// ---------------------------------------------------------------------------
// BlurModel on MI455X (gfx1250, wave32)
//  K1: box-blur (h-pass scalar in LDS, v-pass via v_wmma_f32_16x16x4_f32 with
//      constant banded A) + per-patch histogram on the 0.0005 threshold lattice
//  K2: suffix-sum over histogram -> per-patch threshold (closed form of the
//      reference's bisection walk: min lattice n with count(>t_n) <= 20971)
//  K3: recompute blur in-tile (same WMMA path), threshold, fused 5x5 max-pool
//      + 5x5 min-pool (morphological close) -> d_out
// Tile fills ride the CDNA5 async global->LDS engine (ASYNCcnt): pairs are
// always fully in/out of bounds (even column origins + even image bounds), so
// each pair is one async B64 or a zero store.
// Memory traffic ~192MB total => ~8us at 23.3 TB/s.
// ---------------------------------------------------------------------------

typedef float v2f __attribute__((ext_vector_type(2)));
typedef float v8f __attribute__((ext_vector_type(8)));
typedef int   v2i __attribute__((ext_vector_type(2)));

typedef v2i __attribute__((address_space(1)))* gv2i_p;  // global 64-bit chunk
typedef v2i __attribute__((address_space(3)))* lv2i_p;  // LDS    64-bit chunk

#define IMG 4096
#define NBINS 2002            // lattice n in [-1001, 1000] -> bin = n + 1001
#define HIST_STRIDE 2048
#define COUNT_LIMIT 20971u    // floor(0.08f * 2^18): loop2 stops when count<=this
#define WTAP 0.04f            // 1/25 (f32(0.04) == f32(1.0/25.0))

#if defined(__has_builtin)
#if __has_builtin(__builtin_amdgcn_global_load_async_to_lds_b64)
#define USE_ASYNC_LDS 1
#endif
#endif

__device__ __forceinline__ void fill_pair(const float* __restrict__ x,
                                          float* ldst, int gy, int c0) {
    // loads x[gy*IMG + c0 .. +1] into LDS pair, or zeros if OOB.
    // c0 is even; image bounds are even => pair is never partially OOB.
    if ((unsigned)gy < (unsigned)IMG && (unsigned)c0 < (unsigned)IMG) {
#ifdef USE_ASYNC_LDS
        __builtin_amdgcn_global_load_async_to_lds_b64(
            (gv2i_p)(x + gy * IMG + c0), (lv2i_p)ldst, 0, 0);
#else
        ldst[0] = x[gy * IMG + c0];
        ldst[1] = x[gy * IMG + c0 + 1];
#endif
    } else {
        ldst[0] = 0.0f;
        ldst[1] = 0.0f;
    }
}

__device__ __forceinline__ void wait_async_fill() {
#ifdef USE_ASYNC_LDS
#if __has_builtin(__builtin_amdgcn_s_wait_asynccnt)
    __builtin_amdgcn_s_wait_asynccnt(0);
#else
    asm volatile("s_wait_asynccnt 0" ::: "memory");
#endif
#endif
}

__device__ __forceinline__ int bin_of(float v) {
    // pixel contributes to count(> t_n) for all n <= ceil((v-0.5)*2000)-1
    float q = (v - 0.5f) * 2000.0f;
    int ib = (int)ceilf(q) - 1 + 1001;
    ib = ib < 0 ? 0 : (ib > NBINS - 1 ? NBINS - 1 : ib);
    return ib;
}

// ---- K0: zero the histogram scratch -----------------------------------------
__global__ void k_zero(unsigned* __restrict__ p, int n) {
    int i = blockIdx.x * blockDim.x + threadIdx.x;
    if (i < n) p[i] = 0u;
}

// ---- K1: blur + histogram. 64x64 output tile per 256-thread block ----------
__global__ void __launch_bounds__(256)
k_blur_hist(const float* __restrict__ x, unsigned* __restrict__ gHist) {
    __shared__ float    sIn[68 * 68];   // x tile with halo 2 (row0 <-> Y0-2)
    __shared__ float    sH [68 * 64];   // horizontal 5-sums
    __shared__ unsigned sHist[HIST_STRIDE];

    const int tid = threadIdx.x;
    const int X0 = blockIdx.x * 64, Y0 = blockIdx.y * 64;

    // async fill of the input tile: 68 rows x 34 b64-pairs
    for (int i = tid; i < 68 * 34; i += 256) {
        int r = i / 34, j = i - r * 34;
        fill_pair(x, &sIn[r * 68 + 2 * j], Y0 - 2 + r, X0 - 2 + 2 * j);
    }
    for (int i = tid; i < HIST_STRIDE; i += 256) sHist[i] = 0u;
    wait_async_fill();
    __syncthreads();

    for (int i = tid; i < 68 * 64; i += 256) {
        int r = i / 64, c = i % 64;
        const float* p = &sIn[r * 68 + c];
        sH[i] = p[0] + p[1] + p[2] + p[3] + p[4];
    }
    __syncthreads();

    const int lane = tid & 31, wave = tid >> 5;
    const int m  = lane & 15;         // M row (A) / N col (B,D)
    const int hi = lane >> 4;         // 0: K{0,1}/rows r ; 1: K{2,3}/rows r+8

    for (int t = wave; t < 16; t += 8) {          // 4x4 tiles of 16x16
        const int rb = (t >> 2) * 16, cb = (t & 3) * 16;
        v8f acc = {};
#pragma unroll
        for (int cc = 0; cc < 5; ++cc) {          // K = 20 in 5 chunks of 4
            const int k0 = 4 * cc + 2 * hi;
            v2f a, b;
            a.x = (k0     >= m && k0     <= m + 4) ? WTAP : 0.0f;  // banded
            a.y = (k0 + 1 >= m && k0 + 1 <= m + 4) ? WTAP : 0.0f;
            b.x = sH[(rb + k0    ) * 64 + cb + m];
            b.y = sH[(rb + k0 + 1) * 64 + cb + m];
            acc = __builtin_amdgcn_wmma_f32_16x16x4_f32(
                      false, a, false, b, (short)0, acc, false, false);
        }
#pragma unroll
        for (int r = 0; r < 8; ++r)
            atomicAdd(&sHist[bin_of(acc[r])], 1u);
    }
    __syncthreads();

    const int patch = (Y0 >> 9) * 8 + (X0 >> 9);
    for (int i = tid; i < HIST_STRIDE; i += 256) {
        unsigned u = sHist[i];
        if (u) atomicAdd(&gHist[patch * HIST_STRIDE + i], u);
    }
}

// ---- K2: histogram -> per-patch threshold -----------------------------------
__global__ void k_threshold(const unsigned* __restrict__ gHist,
                            float* __restrict__ th) {
    __shared__ unsigned h[HIST_STRIDE];
    const int p = blockIdx.x, tid = threadIdx.x;
    for (int i = tid; i < HIST_STRIDE; i += blockDim.x)
        h[i] = gHist[p * HIST_STRIDE + i];
    __syncthreads();
    if (tid == 0) {
        unsigned acc = 0;
        int n = 1001;                              // S(1001)=0 always qualifies
        for (int ib = NBINS - 1; ib >= 0; --ib) {  // acc = S(ib-1001) after add
            acc += h[ib];
            if (acc > COUNT_LIMIT) break;
            n = ib - 1001;
        }
        th[p] = 0.5f + (float)n * 0.0005f;
    }
}

// ---- K3: blur + threshold + morphological close, 64x64 out tile -------------
__global__ void __launch_bounds__(256)
k_close(const float* __restrict__ x, const float* __restrict__ th,
        float* __restrict__ out) {
    __shared__ float         sIn[84 * 84];  // x with halo 10 (row0 <-> Y0-10)
    __shared__ float         sH [84 * 80];  // horizontal sums (col0 <-> X0-8)
    __shared__ float         sTh[64];
    __shared__ unsigned char sMax[68 * 68]; // max-pooled (row0 <-> Y0-2)
    unsigned char* sBin = (unsigned char*)sIn;  // 80x80 binary aliases sIn

    const int tid = threadIdx.x;
    const int X0 = blockIdx.x * 64, Y0 = blockIdx.y * 64;

    // async fill of the input tile: 84 rows x 42 b64-pairs
    for (int i = tid; i < 84 * 42; i += 256) {
        int r = i / 42, j = i - r * 42;
        fill_pair(x, &sIn[r * 84 + 2 * j], Y0 - 10 + r, X0 - 10 + 2 * j);
    }
    if (tid < 64) sTh[tid] = th[tid];
    wait_async_fill();
    __syncthreads();

    for (int i = tid; i < 84 * 80; i += 256) {
        int r = i / 80, c = i % 80;
        const float* p = &sIn[r * 84 + c];
        sH[i] = p[0] + p[1] + p[2] + p[3] + p[4];
    }
    __syncthreads();   // sIn reads done -> safe to alias sBin over it

    const int lane = tid & 31, wave = tid >> 5;
    const int m  = lane & 15;
    const int hi = lane >> 4;

    for (int t = wave; t < 25; t += 8) {          // 5x5 tiles cover 80x80
        const int rb = (t / 5) * 16, cb = (t % 5) * 16;
        v8f acc = {};
#pragma unroll
        for (int cc = 0; cc < 5; ++cc) {
            const int k0 = 4 * cc + 2 * hi;
            v2f a, b;
            a.x = (k0     >= m && k0     <= m + 4) ? WTAP : 0.0f;
            a.y = (k0 + 1 >= m && k0 + 1 <= m + 4) ? WTAP : 0.0f;
            b.x = sH[(rb + k0    ) * 80 + cb + m];
            b.y = sH[(rb + k0 + 1) * 80 + cb + m];
            acc = __builtin_amdgcn_wmma_f32_16x16x4_f32(
                      false, a, false, b, (short)0, acc, false, false);
        }
#pragma unroll
        for (int r = 0; r < 8; ++r) {
            const int by = rb + r + 8 * hi, bx = cb + m;  // binary-region coords
            const int gy = Y0 - 8 + by, gx = X0 - 8 + bx;
            unsigned char bit = 0;
            if ((unsigned)gy < (unsigned)IMG && (unsigned)gx < (unsigned)IMG) {
                const int p = (gy >> 9) * 8 + (gx >> 9);
                bit = (acc[r] > sTh[p]) ? 1 : 0;
            }
            sBin[by * 80 + bx] = bit;   // OOB pixels -> 0 (safe for max-pool)
        }
    }
    __syncthreads();

    // 5x5 max-pool onto 68x68 (halo 2 for the min-pool); OOB centers -> 1
    for (int i = tid; i < 68 * 68; i += 256) {
        int r = i / 68, c = i % 68;
        int my = Y0 - 2 + r, mx = X0 - 2 + c;
        unsigned char v = 1;
        if ((unsigned)my < (unsigned)IMG && (unsigned)mx < (unsigned)IMG) {
            v = 0;
#pragma unroll
            for (int k = 0; k < 5; ++k)
#pragma unroll
                for (int kk = 0; kk < 5; ++kk)
                    v |= sBin[(r + 4 + k) * 80 + (c + 4 + kk)];
        }
        sMax[i] = v;
    }
    __syncthreads();

    // 5x5 min-pool -> final 64x64 output
    for (int i = tid; i < 64 * 64; i += 256) {
        int r = i / 64, c = i % 64;
        unsigned char v = 1;
#pragma unroll
        for (int k = 0; k < 5; ++k)
#pragma unroll
            for (int kk = 0; kk < 5; ++kk)
                v &= sMax[(r + k) * 68 + (c + kk)];
        out[(Y0 + r) * IMG + (X0 + c)] = (float)v;
    }
}

// ---------------------------------------------------------------------------
extern "C" void kernel_launch(void* const* d_in, const int* in_sizes, int n_in,
                              void* d_out, int out_size, void* d_ws, size_t ws_size,
                              hipStream_t stream) {
    const float* x = (const float*)d_in[0];
    // d_in[1] = blur_k: constant 1/25 box filter, folded into WTAP
    float* out = (float*)d_out;

    unsigned* hist = (unsigned*)d_ws;                       // 64 * 2048 u32
    float*    th   = (float*)((char*)d_ws + 64 * HIST_STRIDE * sizeof(unsigned));

    const int nhist = 64 * HIST_STRIDE;
    k_zero<<<(nhist + 255) / 256, 256, 0, stream>>>(hist, nhist);

    dim3 grid(IMG / 64, IMG / 64);
    k_blur_hist<<<grid, 256, 0, stream>>>(x, hist);
    k_threshold<<<64, 256, 0, stream>>>(hist, th);
    k_close<<<grid, 256, 0, stream>>>(x, th, out);
}